// GramSchmidt_51393578664641
// MI455X (gfx1250) — compile-verified
//
#include <hip/hip_runtime.h>
#include <hip/hip_bf16.h>

// Real-interleaved view: Q is M x 128 row-major floats (col 2c = Re, 2c+1 = Im).
#define RC 128
#define RCOLS 64
#define KCHUNK 32  // K-rows staged per async LDS chunk (32*128*4B = 16 KB)

typedef float v2f __attribute__((ext_vector_type(2)));
typedef float v8f __attribute__((ext_vector_type(8)));

__device__ __forceinline__ float wave_reduce_add(float v) {
#pragma unroll
  for (int off = 16; off > 0; off >>= 1) v += __shfl_xor(v, off, 32);
  return v;
}

__device__ __forceinline__ uint32_t lds_addr_of(const void* p) {
  return (uint32_t)(uintptr_t)p;
}

__device__ __forceinline__ void async_load_b128(uint32_t lds_off,
                                                const void* gaddr) {
  asm volatile("global_load_async_to_lds_b128 %0, %1, off" ::"v"(lds_off),
               "v"(gaddr)
               : "memory");
}

__global__ void zero_f32_kernel(float* __restrict__ p, int n) {
  for (int i = blockIdx.x * blockDim.x + threadIdx.x; i < n;
       i += gridDim.x * blockDim.x)
    p[i] = 0.f;
}

// ---------------- Gram-Schmidt prologue (CGS; == MGS in exact arithmetic) ----
// coeff[2j],coeff[2j+1] = Re/Im of <q_j, x_i>, j-loop tiled by 8 so x_i is
// streamed once per 8 previous columns.
__global__ void cgs_coeff_kernel(const float* __restrict__ X,
                                 const float* __restrict__ Q,
                                 float* __restrict__ coeff, int i, int M) {
  const int lane = threadIdx.x & 31;
  for (int jt = 0; jt < i; jt += 8) {
    const int je = min(jt + 8, i);
    float re[8] = {}, im[8] = {};
    for (int m = blockIdx.x * blockDim.x + threadIdx.x; m < M;
         m += gridDim.x * blockDim.x) {
      const float* qrow = Q + (size_t)m * RC;
      const float* xp = X + (size_t)m * RC + 2 * i;
      const float xr = xp[0], xi = xp[1];
      for (int j = jt; j < je; ++j) {
        const float qr = qrow[2 * j], qi = qrow[2 * j + 1];
        re[j - jt] += qr * xr + qi * xi;  // Re(conj(qj) * x)
        im[j - jt] += qr * xi - qi * xr;  // Im(conj(qj) * x)
      }
    }
    for (int j = jt; j < je; ++j) {
      const float r = wave_reduce_add(re[j - jt]);
      const float s = wave_reduce_add(im[j - jt]);
      if (lane == 0) {
        atomicAdd(&coeff[2 * j], r);
        atomicAdd(&coeff[2 * j + 1], s);
      }
    }
  }
}

// q_i = x_i - sum_j coeff_j * q_j ; accumulate |q_i|^2 into nrm[0].
__global__ void cgs_update_kernel(const float* __restrict__ X,
                                  float* __restrict__ Q,
                                  const float* __restrict__ coeff,
                                  float* __restrict__ nrm, int i, int M) {
  const int lane = threadIdx.x & 31;
  float acc = 0.f;
  for (int m = blockIdx.x * blockDim.x + threadIdx.x; m < M;
       m += gridDim.x * blockDim.x) {
    const float* xp = X + (size_t)m * RC + 2 * i;
    __builtin_prefetch(xp + (size_t)gridDim.x * blockDim.x * RC, 0, 1);
    float xr = xp[0], xi = xp[1];
    for (int j = 0; j < i; ++j) {
      const float cr = coeff[2 * j], ci = coeff[2 * j + 1];
      const float qr = Q[(size_t)m * RC + 2 * j];
      const float qi = Q[(size_t)m * RC + 2 * j + 1];
      xr -= cr * qr - ci * qi;  // x -= c * q_j (complex)
      xi -= cr * qi + ci * qr;
    }
    Q[(size_t)m * RC + 2 * i] = xr;
    Q[(size_t)m * RC + 2 * i + 1] = xi;
    acc += xr * xr + xi * xi;
  }
  acc = wave_reduce_add(acc);
  if (lane == 0) atomicAdd(nrm, acc);
}

__global__ void col_scale_kernel(float* __restrict__ Q,
                                 const float* __restrict__ nrm, int i, int M) {
  const float s = rsqrtf(nrm[0] + 1e-10f);
  for (int m = blockIdx.x * blockDim.x + threadIdx.x; m < M;
       m += gridDim.x * blockDim.x) {
    Q[(size_t)m * RC + 2 * i] *= s;
    Q[(size_t)m * RC + 2 * i + 1] *= s;
  }
}

// ---------------- Björck refinement: WMMA f32 path ---------------------------
// S = X^T X (128x128 real SYRK, K = M). K-chunks of 32 rows are DMA'd into LDS
// with global_load_async_to_lds_b128 (ASYNCcnt), double buffered against the
// WMMA K-loop. Wave w owns S tile-row w x 8 tile-cols.
__global__ void gram_syrk_wmma_kernel(const float* __restrict__ X,
                                      float* __restrict__ S, int M) {
  __shared__ float Xs[2][KCHUNK * RC];  // 2 x 16 KB

  const int lane = threadIdx.x & 31;
  const int wave = threadIdx.x >> 5;     // 0..7 -> S tile row
  const int half = (lane < 16) ? 0 : 2;  // K sub-offset for this lane
  const int lcol = lane & 15;

  int span = (M + gridDim.x - 1) / gridDim.x;
  span = (span + KCHUNK - 1) & ~(KCHUNK - 1);
  const int k0 = blockIdx.x * span;
  const int k1 = min(k0 + span, M);
  const int nch = (k1 > k0) ? (k1 - k0) / KCHUNK : 0;  // full chunks

  v8f acc[8] = {};
  const int colA = wave * 16 + lcol;

  // Issue async DMA of chunk c into buffer b: 4 x b128 per thread.
  auto stage = [&](int c, int b) {
    const float* src = X + (size_t)(k0 + c * KCHUNK) * RC;
    const uint32_t dst = lds_addr_of(&Xs[b][0]);
#pragma unroll
    for (int t = 0; t < (KCHUNK * RC) / (4 * 256); ++t) {  // 4 iterations
      const int f = threadIdx.x + t * 256;                 // float4 index
      async_load_b128(dst + f * 16, src + (size_t)f * 4);
    }
  };

  if (nch > 0) stage(0, 0);
  for (int c = 0; c < nch; ++c) {
    const int b = c & 1;
    if (c + 1 < nch) {
      stage(c + 1, b ^ 1);  // overlap DMA of next chunk with WMMA on this one
      asm volatile("s_wait_asynccnt 0x4" ::: "memory");
    } else {
      asm volatile("s_wait_asynccnt 0x0" ::: "memory");
    }
    __syncthreads();
    const float* xs = &Xs[b][0];
#pragma unroll
    for (int kk = 0; kk < KCHUNK; kk += 4) {
      v2f a;
      a.x = xs[(kk + half) * RC + colA];
      a.y = xs[(kk + half + 1) * RC + colA];
#pragma unroll
      for (int n = 0; n < 8; ++n) {
        const int colB = n * 16 + lcol;
        v2f bf;
        bf.x = xs[(kk + half) * RC + colB];
        bf.y = xs[(kk + half + 1) * RC + colB];
        acc[n] = __builtin_amdgcn_wmma_f32_16x16x4_f32(false, a, false, bf,
                                                       (short)0, acc[n], false,
                                                       false);
      }
    }
    __syncthreads();
  }

  // Scalar-global tail for K ranges not covering full chunks.
  for (int k = k0 + nch * KCHUNK; k + 3 < k1; k += 4) {
    const float* r0 = X + (size_t)(k + half) * RC;
    const float* r1 = r0 + RC;
    v2f a;
    a.x = r0[colA];
    a.y = r1[colA];
#pragma unroll
    for (int n = 0; n < 8; ++n) {
      const int colB = n * 16 + lcol;
      v2f bf;
      bf.x = r0[colB];
      bf.y = r1[colB];
      acc[n] = __builtin_amdgcn_wmma_f32_16x16x4_f32(false, a, false, bf,
                                                     (short)0, acc[n], false,
                                                     false);
    }
  }

  const int rowBase = wave * 16 + ((lane < 16) ? 0 : 8);
#pragma unroll
  for (int n = 0; n < 8; ++n) {
#pragma unroll
    for (int r = 0; r < 8; ++r) {
      atomicAdd(&S[(size_t)(rowBase + r) * RC + n * 16 + lcol], acc[n][r]);
    }
  }
}

// Build real-expanded correction F (128x128) from S:
//   G = Q^H Q, E = 0.4*(G - I);  F[2a,2b]=Re E, F[2a+1,2b+1]=Re E,
//   F[2a,2b+1]=Im E, F[2a+1,2b]=-Im E.
__global__ void build_F_kernel(const float* __restrict__ S,
                               float* __restrict__ F) {
  const int idx = blockIdx.x * blockDim.x + threadIdx.x;
  if (idx >= RCOLS * RCOLS) return;
  const int a = idx >> 6, b = idx & 63;
  const float gre = S[(size_t)(2 * a) * RC + 2 * b] +
                    S[(size_t)(2 * a + 1) * RC + 2 * b + 1];
  const float gim = S[(size_t)(2 * a) * RC + 2 * b + 1] -
                    S[(size_t)(2 * a + 1) * RC + 2 * b];
  const float ere = 0.4f * (gre - (a == b ? 1.f : 0.f));
  const float eim = 0.4f * gim;
  F[(size_t)(2 * a) * RC + 2 * b] = ere;
  F[(size_t)(2 * a + 1) * RC + 2 * b + 1] = ere;
  F[(size_t)(2 * a) * RC + 2 * b + 1] = eim;
  F[(size_t)(2 * a + 1) * RC + 2 * b] = -eim;
}

// Q = Q - Q @ F  (M x 128)(128 x 128), WMMA K-loop over 128; F is DMA'd to LDS
// with async b128 loads. In-place safe: each wave's A-reads touch only its own
// 16 rows.
__global__ void bjorck_correct_wmma_kernel(float* __restrict__ Q,
                                           const float* __restrict__ F,
                                           int M) {
  __shared__ float Fs[RC * RC];  // 64 KB of the 320 KB WGP LDS
  {
    const uint32_t dst = lds_addr_of(&Fs[0]);
#pragma unroll
    for (int t = 0; t < (RC * RC) / (4 * 256); ++t) {  // 16 iterations
      const int f = threadIdx.x + t * 256;             // float4 index
      async_load_b128(dst + f * 16, F + (size_t)f * 4);
    }
    asm volatile("s_wait_asynccnt 0x0" ::: "memory");
  }
  __syncthreads();

  const int lane = threadIdx.x & 31;
  const int wave = threadIdx.x >> 5;
  const int half = (lane < 16) ? 0 : 2;
  const int lcol = lane & 15;

  const int row0 = (blockIdx.x * 8 + wave) * 16;
  const int rowA = min(row0 + lcol, M - 1);
  const float* qrow = Q + (size_t)rowA * RC;

  v8f acc[8] = {};
  for (int k = 0; k < RC; k += 4) {
    v2f a;
    a.x = qrow[k + half];
    a.y = qrow[k + half + 1];
#pragma unroll
    for (int n = 0; n < 8; ++n) {
      const int colB = n * 16 + lcol;
      v2f bf;
      bf.x = Fs[(k + half) * RC + colB];
      bf.y = Fs[(k + half + 1) * RC + colB];
      acc[n] = __builtin_amdgcn_wmma_f32_16x16x4_f32(false, a, false, bf,
                                                     (short)0, acc[n], false,
                                                     false);
    }
  }

#pragma unroll
  for (int r = 0; r < 8; ++r) {
    const int row = row0 + r + ((lane < 16) ? 0 : 8);
    if (row < M) {
      float* out = Q + (size_t)row * RC;
#pragma unroll
      for (int n = 0; n < 8; ++n) out[n * 16 + lcol] -= acc[n][r];
    }
  }
}

__global__ void colnorm_kernel(const float* __restrict__ Q,
                               float* __restrict__ nrm, int M) {
  const int c = blockIdx.y;
  const int lane = threadIdx.x & 31;
  float acc = 0.f;
  for (int m = blockIdx.x * blockDim.x + threadIdx.x; m < M;
       m += gridDim.x * blockDim.x) {
    const float qr = Q[(size_t)m * RC + 2 * c];
    const float qi = Q[(size_t)m * RC + 2 * c + 1];
    acc += qr * qr + qi * qi;
  }
  acc = wave_reduce_add(acc);
  if (lane == 0) atomicAdd(&nrm[c], acc);
}

__global__ void scale_all_kernel(float* __restrict__ Q,
                                 const float* __restrict__ nrm, int n) {
  for (int i = blockIdx.x * blockDim.x + threadIdx.x; i < n;
       i += gridDim.x * blockDim.x) {
    const int c = (i & (RC - 1)) >> 1;
    Q[i] *= rsqrtf(nrm[c] + 1e-10f);
  }
}

extern "C" void kernel_launch(void* const* d_in, const int* in_sizes, int n_in,
                              void* d_out, int out_size, void* d_ws,
                              size_t ws_size, hipStream_t stream) {
  (void)n_in; (void)out_size; (void)ws_size;
  const float* X = (const float*)d_in[0];
  float* Q = (float*)d_out;
  const int M = in_sizes[0] / RC;

  float* S = (float*)d_ws;     // 128*128 floats
  float* F = S + RC * RC;      // 128*128 floats
  float* coeff = F + RC * RC;  // 128 floats
  float* nrm = coeff + RC;     // 64 floats

  // ---- Gram-Schmidt over 64 columns ----
  for (int i = 0; i < RCOLS; ++i) {
    if (i > 0) {
      zero_f32_kernel<<<1, 256, 0, stream>>>(coeff, 2 * i);
      cgs_coeff_kernel<<<256, 256, 0, stream>>>(X, Q, coeff, i, M);
    }
    zero_f32_kernel<<<1, 32, 0, stream>>>(nrm, 1);
    cgs_update_kernel<<<256, 256, 0, stream>>>(X, Q, coeff, nrm, i, M);
    col_scale_kernel<<<256, 256, 0, stream>>>(Q, nrm, i, M);
  }

  // ---- 3 Björck refinement sweeps (WMMA GEMMs) ----
  const int corrBlocks = (M + 127) / 128;  // 8 waves * 16 rows per block
  for (int it = 0; it < 3; ++it) {
    zero_f32_kernel<<<64, 256, 0, stream>>>(S, RC * RC);
    gram_syrk_wmma_kernel<<<512, 256, 0, stream>>>(Q, S, M);
    build_F_kernel<<<16, 256, 0, stream>>>(S, F);
    bjorck_correct_wmma_kernel<<<corrBlocks, 256, 0, stream>>>(Q, F, M);
    zero_f32_kernel<<<1, 64, 0, stream>>>(nrm, RCOLS);
    colnorm_kernel<<<dim3(32, RCOLS), 256, 0, stream>>>(Q, nrm, M);
    scale_all_kernel<<<512, 256, 0, stream>>>(Q, nrm, M * RC);
  }
}